// GraphSAGE_51651276702105
// MI455X (gfx1250) — compile-verified
//
#include <hip/hip_runtime.h>
#include <hip/hip_bf16.h>

typedef __attribute__((ext_vector_type(2))) float v2f;
typedef __attribute__((ext_vector_type(8))) float v8f;

#define D 128          // feature dim
#define K2 256         // concat dim

// ---------------------------------------------------------------------------
// Scatter-add: one wave per edge. Lane l handles features [4l, 4l+4).
// sum[tgt] += x[src]; optionally deg[tgt] += 1 (edges identical both layers).
// ---------------------------------------------------------------------------
__global__ __launch_bounds__(256) void sage_scatter(
    const float* __restrict__ x,
    const long long* __restrict__ src,
    const long long* __restrict__ tgt,
    float* __restrict__ sum,
    float* __restrict__ deg,
    int E, int addDeg)
{
    int wave = (blockIdx.x * blockDim.x + threadIdx.x) >> 5;
    int lane = threadIdx.x & 31;
    if (wave >= E) return;

    long long s = src[wave];
    long long t = tgt[wave];

    const float4 v = *(const float4*)(x + s * (long long)D + lane * 4);
    float* dst = sum + t * (long long)D + lane * 4;
    unsafeAtomicAdd(dst + 0, v.x);
    unsafeAtomicAdd(dst + 1, v.y);
    unsafeAtomicAdd(dst + 2, v.z);
    unsafeAtomicAdd(dst + 3, v.w);
    if (addDeg && lane == 0) unsafeAtomicAdd(deg + t, 1.0f);
}

// ---------------------------------------------------------------------------
// Fused layer: out[16 rows x 16 cols per wave] =
//   relu( [x | sum/max(deg,1)] @ W + b )
// using V_WMMA_F32_16X16X4_F32, K = 256 in 64 chained steps.
// Block = 128 threads = 4 waves -> 16 rows x 64 cols per block.
// Grid  = (N/16, 2) to cover all 128 output columns.
// ---------------------------------------------------------------------------
__global__ __launch_bounds__(128) void sage_layer_wmma(
    const float* __restrict__ xin,   // [N, 128]
    const float* __restrict__ sum,   // [N, 128]
    const float* __restrict__ deg,   // [N]
    const float* __restrict__ W,     // [256, 128] row-major (in, out)
    const float* __restrict__ bias,  // [128]
    float* __restrict__ out,         // [N, 128]
    int N)
{
    const int lane  = threadIdx.x & 31;
    const int waveI = threadIdx.x >> 5;

    const int row0 = blockIdx.x * 16;
    const int n0   = blockIdx.y * 64 + waveI * 16;
    if (row0 >= N) return;

    const int ar    = lane & 15;        // A-fragment row (M)
    const int khalf = (lane >> 4) * 2;  // lanes 16-31 hold K+2,K+3
    const int col   = n0 + (lane & 15); // B/D column (N)

    const float* xrow = xin + (size_t)(row0 + ar) * D;
    const float* srow = sum + (size_t)(row0 + ar) * D;
    const float  inv  = 1.0f / fmaxf(deg[row0 + ar], 1.0f);

    v8f acc = {};

    // ---- K = 0..127 : x part of the concat -------------------------------
    #pragma unroll 4
    for (int k0 = 0; k0 < D; k0 += 4) {
        const int k = k0 + khalf;
        v2f a; a.x = xrow[k];
               a.y = xrow[k + 1];
        v2f b; b.x = W[(size_t)k * D + col];
               b.y = W[(size_t)(k + 1) * D + col];
        acc = __builtin_amdgcn_wmma_f32_16x16x4_f32(
            false, a, false, b, (short)0, acc, false, false);
    }
    // ---- K = 128..255 : mean-aggregated part -----------------------------
    #pragma unroll 4
    for (int k0 = 0; k0 < D; k0 += 4) {
        const int k  = k0 + khalf;
        const int kk = D + k;
        v2f a; a.x = srow[k]     * inv;
               a.y = srow[k + 1] * inv;
        v2f b; b.x = W[(size_t)kk * D + col];
               b.y = W[(size_t)(kk + 1) * D + col];
        acc = __builtin_amdgcn_wmma_f32_16x16x4_f32(
            false, a, false, b, (short)0, acc, false, false);
    }

    // ---- bias + ReLU + store ---------------------------------------------
    const float bv    = bias[col];
    const int   mbase = (lane >> 4) * 8;   // lanes 16-31 hold rows M=8..15
    #pragma unroll
    for (int r = 0; r < 8; ++r) {
        const int m = row0 + mbase + r;
        float v = acc[r] + bv;
        v = fmaxf(v, 0.0f);
        out[(size_t)m * D + col] = v;
    }
}

// ---------------------------------------------------------------------------
// Host driver. Inputs: x[N,128], edge_index[2,E] (int64), W1[256,128],
// b1[128], W2[256,128], b2[128]. Output: [N,128] float32.
// ---------------------------------------------------------------------------
extern "C" void kernel_launch(void* const* d_in, const int* in_sizes, int n_in,
                              void* d_out, int out_size, void* d_ws, size_t ws_size,
                              hipStream_t stream)
{
    const float*      x  = (const float*)d_in[0];
    const long long*  ei = (const long long*)d_in[1];
    const float*      W1 = (const float*)d_in[2];
    const float*      b1 = (const float*)d_in[3];
    const float*      W2 = (const float*)d_in[4];
    const float*      b2 = (const float*)d_in[5];
    float*            out = (float*)d_out;

    const int N = in_sizes[0] / D;       // 50000
    const int E = in_sizes[1] / 2;       // 800000
    const long long* src = ei;
    const long long* tgt = ei + E;

    // Workspace layout: sum [N*128] | deg [N] | h [N*128]
    float* sum = (float*)d_ws;
    float* deg = sum + (size_t)N * D;
    float* h   = deg + N;

    const int scatterBlocks = (E + 7) / 8;          // 8 waves (edges) / block
    const dim3 layerGrid((N + 15) / 16, 2);

    // ---- Layer 1 ----
    hipMemsetAsync(sum, 0, ((size_t)N * D + N) * sizeof(float), stream);
    sage_scatter<<<scatterBlocks, 256, 0, stream>>>(x, src, tgt, sum, deg, E, 1);
    sage_layer_wmma<<<layerGrid, 128, 0, stream>>>(x, sum, deg, W1, b1, h, N);

    // ---- Layer 2 (deg unchanged: same edge list) ----
    hipMemsetAsync(sum, 0, (size_t)N * D * sizeof(float), stream);
    sage_scatter<<<scatterBlocks, 256, 0, stream>>>(h, src, tgt, sum, deg, E, 0);
    sage_layer_wmma<<<layerGrid, 128, 0, stream>>>(h, sum, deg, W2, b2, out, N);
}